// Get_gradient_62165356642501
// MI455X (gfx1250) — compile-verified
//
#include <hip/hip_runtime.h>

// Sobel gradient magnitude, f32, zero padding.
// x: [16,3,544,960] -> 48 planes of 544x960.
// out[h,w] = sqrt((x[h+1,w]-x[h-1,w])^2 + (x[h,w+1]-x[h,w-1])^2 + 1e-6)

#define H_DIM 544
#define W_DIM 960
#define N_PLANES 48            // 16 * 3
#define TH 16                  // output tile rows
#define TW 128                 // output tile cols
#define HALO 4                 // staged halo floats each side (1 needed, 4 keeps 16B alignment)
#define SW (TW + 2 * HALO)     // 136 staged floats per row
#define SH (TH + 2)            // 18 staged rows
#define SW4 (SW / 4)           // 34 float4 per staged row
#define NTHREADS 256
#define EPS_VAL 1e-6f

typedef float f32x4 __attribute__((ext_vector_type(4)));
typedef int   v4i   __attribute__((ext_vector_type(4)));

// ---- CDNA5 async global->LDS copy (ASYNCcnt path) ----
__device__ __forceinline__ void async_copy_b128(const float* gp, float* lp) {
#if defined(__gfx1250__) && __has_builtin(__builtin_amdgcn_global_load_async_to_lds_b128)
    __builtin_amdgcn_global_load_async_to_lds_b128(
        (__attribute__((address_space(1))) v4i*)gp,
        (__attribute__((address_space(3))) v4i*)lp,
        0 /*imm offset*/, 0 /*cpol*/);
#else
    // Fallback: VDST = LDS byte address (low 32 bits of generic LDS pointer),
    // VADDR = 64-bit global address.
    unsigned lds_off = (unsigned)(unsigned long long)lp;
    asm volatile("global_load_async_to_lds_b128 %0, %1, off"
                 :: "v"(lds_off), "v"(gp)
                 : "memory");
#endif
}

__device__ __forceinline__ void wait_async_zero() {
#if defined(__gfx1250__) && __has_builtin(__builtin_amdgcn_s_wait_asynccnt)
    __builtin_amdgcn_s_wait_asynccnt(0);
#else
    asm volatile("s_wait_asynccnt 0" ::: "memory");
#endif
}

__global__ __launch_bounds__(NTHREADS)
void sobel_mag_kernel(const float* __restrict__ x, float* __restrict__ out) {
    __shared__ __align__(16) float tile[SH * SW];   // 18*136*4 = 9792 B

    const int tid = threadIdx.x;
    const int w0  = blockIdx.x * TW;     // 0..896
    const int h0  = blockIdx.y * TH;     // 0..528
    const long plane = (long)blockIdx.z * (long)(H_DIM * W_DIM);
    const float* src = x + plane;
    float*       dst = out + plane;

    // ---------- stage 18 x 136 floats: zero OOB halo, async-load the rest ----------
    // 18 rows * 34 float4 = 612 b128 transfers per block.
    for (int i = tid; i < SH * SW4; i += NTHREADS) {
        const int r  = i / SW4;                 // staged row 0..17
        const int c4 = i % SW4;                 // float4 column 0..33
        const int gh = h0 - 1 + r;              // global row
        const int gw = w0 - HALO + c4 * 4;      // global col of first float
        float* lp = &tile[r * SW + c4 * 4];
        if (gh >= 0 && gh < H_DIM && gw >= 0 && gw + 4 <= W_DIM) {
            async_copy_b128(src + (long)gh * W_DIM + gw, lp);
        } else {
            f32x4 z = {0.f, 0.f, 0.f, 0.f};
            *(f32x4*)lp = z;                    // aligned ds_store_b128
        }
    }
    wait_async_zero();       // this wave's async loads landed in LDS
    __syncthreads();         // all waves' loads + zero stores visible

    // ---------- compute: each thread -> two float4 row segments ----------
    const int lane_col = tid & 31;              // 0..31 -> 4-float column group
    const int row_base = tid >> 5;              // 0..7
    const int lc   = HALO + lane_col * 4;       // staged col of first output float
    const int wcol = w0 + lane_col * 4;         // global output col

#pragma unroll
    for (int rr = 0; rr < 2; ++rr) {
        const int r  = row_base + rr * 8;       // output row within tile, 0..15
        const int lr = r + 1;                   // staged row

        const float* rowm = &tile[(lr - 1) * SW + lc];
        const float* rowc = &tile[(lr    ) * SW + lc];
        const float* rowp = &tile[(lr + 1) * SW + lc];

        f32x4 up  = *(const f32x4*)rowm;        // aligned ds_load_b128
        f32x4 dn  = *(const f32x4*)rowp;
        f32x4 mid = *(const f32x4*)rowc;
        float lm1 = rowc[-1];                   // shifted horizontal neighbors
        float lp4 = rowc[4];

        f32x4 gv = dn - up;                     // x[h+1]-x[h-1]
        f32x4 gh4;
        gh4.x = mid.y - lm1;                    // x[w+1]-x[w-1]
        gh4.y = mid.z - mid.x;
        gh4.z = mid.w - mid.y;
        gh4.w = lp4   - mid.z;

        f32x4 res;
        res.x = __builtin_sqrtf(gv.x * gv.x + gh4.x * gh4.x + EPS_VAL);
        res.y = __builtin_sqrtf(gv.y * gv.y + gh4.y * gh4.y + EPS_VAL);
        res.z = __builtin_sqrtf(gv.z * gv.z + gh4.z * gh4.z + EPS_VAL);
        res.w = __builtin_sqrtf(gv.w * gv.w + gh4.w * gh4.w + EPS_VAL);

        if (wcol < W_DIM) {                     // last column tile is partial (960 = 7.5*128)
            f32x4* op = (f32x4*)(dst + (long)(h0 + r) * W_DIM + wcol);
            __builtin_nontemporal_store(res, op);   // write-once output: NT store
        }
    }
}

extern "C" void kernel_launch(void* const* d_in, const int* in_sizes, int n_in,
                              void* d_out, int out_size, void* d_ws, size_t ws_size,
                              hipStream_t stream) {
    (void)in_sizes; (void)n_in; (void)out_size; (void)d_ws; (void)ws_size;
    const float* x = (const float*)d_in[0];
    float* out = (float*)d_out;

    dim3 grid((W_DIM + TW - 1) / TW,   // 8  (last tile partial)
              H_DIM / TH,              // 34
              N_PLANES);               // 48
    dim3 block(NTHREADS);
    sobel_mag_kernel<<<grid, block, 0, stream>>>(x, out);
}